// LengthRegulator_20890720928379
// MI455X (gfx1250) — compile-verified
//
#include <hip/hip_runtime.h>
#include <stdint.h>

// LengthRegulator for MI455X (gfx1250).
// Pure data-movement kernel: ~113 MB stores + <=64 MB (L2-resident) loads.
// Roofline @ 23.3 TB/s HBM  =>  ~7 us.  No FLOPs -> WMMA inapplicable;
// the CDNA5 path used here is the async global<->LDS DMA pipeline
// (GLOBAL_LOAD_ASYNC_TO_LDS_B128 / GLOBAL_STORE_ASYNC_FROM_LDS_B128,
// tracked with ASYNCcnt), which moves each 2048-byte row without the data
// ever occupying VGPRs.

#define B_DIM 64
#define T_DIM 512
#define HIDDEN 512
#define TARGET_FRAMES 861              // 22050*10/256
#define ROW_BYTES (HIDDEN * 4)         // 2048 bytes per row
#define WAVES_PER_BLOCK 8
#define GATHER_THREADS (WAVES_PER_BLOCK * 32)

// ---------------------------------------------------------------------------
// Kernel 1: per-batch duration scan + inverted searchsorted (frame -> token).
// idx_ws[b][f] = token owning frame f, or -1 if f >= total (zero-fill).
// ---------------------------------------------------------------------------
__global__ __launch_bounds__(T_DIM) void lr_build_index(
    const float* __restrict__ pd, int* __restrict__ idx_ws)
{
    __shared__ int ends[T_DIM];
    __shared__ int idxbuf[TARGET_FRAMES];

    const int b = blockIdx.x;
    const int t = threadIdx.x;

    // d = round(clip(pd, 1, 10)); rintf = round-to-nearest-even == jnp.round
    float v = pd[(size_t)b * T_DIM + t];
    v = fminf(fmaxf(v, 1.0f), 10.0f);
    int d = (int)rintf(v);

    // Hillis-Steele inclusive scan over 512 elements
    ends[t] = d;
    __syncthreads();
    for (int off = 1; off < T_DIM; off <<= 1) {
        int add = (t >= off) ? ends[t - off] : 0;
        __syncthreads();
        ends[t] += add;
        __syncthreads();
    }

    // init frame->token map to -1 (frames beyond total stay -1)
    for (int f = t; f < TARGET_FRAMES; f += T_DIM) idxbuf[f] = -1;
    __syncthreads();

    // token t covers frames [ends[t-1], ends[t])  (durations >= 1 => disjoint)
    int start = (t == 0) ? 0 : ends[t - 1];
    int stop  = ends[t];
    if (stop > TARGET_FRAMES) stop = TARGET_FRAMES;
    for (int f = start; f < stop; ++f) idxbuf[f] = t;
    __syncthreads();

    for (int f = t; f < TARGET_FRAMES; f += T_DIM)
        idx_ws[(size_t)b * TARGET_FRAMES + f] = idxbuf[f];
}

// ---------------------------------------------------------------------------
// Kernel 2: row gather. One wave (32 lanes) per output row.
// Valid row: async DMA global -> LDS -> global, 4 x (32 lanes x 16B) = 2048B.
// Invalid row (-1): coalesced float4 zero stores.
// ---------------------------------------------------------------------------
__global__ __launch_bounds__(GATHER_THREADS) void lr_gather(
    const float* __restrict__ batch, const int* __restrict__ idx_ws,
    float* __restrict__ out)
{
    __shared__ __align__(16) unsigned char stage[WAVES_PER_BLOCK * ROW_BYTES];

    const int lane = threadIdx.x & 31;
    const int wave = threadIdx.x >> 5;
    const int row  = blockIdx.x * WAVES_PER_BLOCK + wave;
    if (row >= B_DIM * TARGET_FRAMES) return;   // grid divides evenly; kept for safety

    const int b   = row / TARGET_FRAMES;
    const int tok = idx_ws[row];                // wave-uniform

    float* orow = out + (size_t)row * HIDDEN;

    if (tok < 0) {
        // frame >= total: zero row, full-width 128-bit stores
        float4  z  = make_float4(0.f, 0.f, 0.f, 0.f);
        float4* o4 = reinterpret_cast<float4*>(orow);
#pragma unroll
        for (int k = 0; k < 4; ++k) o4[lane + 32 * k] = z;
        return;
    }

    const float* srow = batch + ((size_t)b * T_DIM + tok) * HIDDEN;

    uint64_t gsrc = (uint64_t)(uintptr_t)srow + (uint32_t)(lane * 16);
    uint64_t gdst = (uint64_t)(uintptr_t)orow + (uint32_t)(lane * 16);
    // low 32 bits of the generic pointer == LDS byte address (ISA 10.2 aperture)
    uint32_t lds  = (uint32_t)(uintptr_t)&stage[wave * ROW_BYTES + lane * 16];

    // global -> LDS, 2048 bytes per wave; offset: applies to both addresses
    asm volatile(
        "global_load_async_to_lds_b128 %0, %1, off\n\t"
        "global_load_async_to_lds_b128 %0, %1, off offset:512\n\t"
        "global_load_async_to_lds_b128 %0, %1, off offset:1024\n\t"
        "global_load_async_to_lds_b128 %0, %1, off offset:1536"
        :: "v"(lds), "v"(gsrc) : "memory");

    // loads vs stores complete out of order on ASYNCcnt -> must drain loads
    asm volatile("s_wait_asynccnt 0" ::: "memory");

    // LDS -> global
    asm volatile(
        "global_store_async_from_lds_b128 %0, %1, off\n\t"
        "global_store_async_from_lds_b128 %0, %1, off offset:512\n\t"
        "global_store_async_from_lds_b128 %0, %1, off offset:1024\n\t"
        "global_store_async_from_lds_b128 %0, %1, off offset:1536"
        :: "v"(gdst), "v"(lds) : "memory");

    asm volatile("s_wait_asynccnt 0" ::: "memory");
}

// ---------------------------------------------------------------------------
extern "C" void kernel_launch(void* const* d_in, const int* in_sizes, int n_in,
                              void* d_out, int out_size, void* d_ws, size_t ws_size,
                              hipStream_t stream) {
    (void)in_sizes; (void)n_in; (void)out_size; (void)ws_size;

    const float* batch = (const float*)d_in[0];   // (64, 512, 512) fp32
    const float* pd    = (const float*)d_in[1];   // (64, 512, 1)  fp32
    float* out   = (float*)d_out;                 // (64, 861, 512) fp32
    int* idx_ws  = (int*)d_ws;                    // 64*861 ints = 220 KB scratch

    lr_build_index<<<B_DIM, T_DIM, 0, stream>>>(pd, idx_ws);

    const int rows   = B_DIM * TARGET_FRAMES;                    // 55104
    const int blocks = (rows + WAVES_PER_BLOCK - 1) / WAVES_PER_BLOCK; // 6888
    lr_gather<<<blocks, GATHER_THREADS, 0, stream>>>(batch, idx_ws, out);
}